// DualTimeConstantHighPassMixAdaptation_21844203667494
// MI455X (gfx1250) — compile-verified
//
#include <hip/hip_runtime.h>
#include <hip/hip_bf16.h>

// MI455X / gfx1250, wave32. Chunked-scan EMA:
//   P1: per-(channel,chunk) weighted sums (coalesced streaming, 64000 waves)
//   P2: per-channel serial composition over 125 chunk summaries (512 threads)
//   P3: in-chunk reconstruction as triangular-Toeplitz matmul on the WMMA pipe
//       (v_wmma_f32_16x16x4_f32, 10 per 16t x 16ch tile) + fused output math.
// Branch-free tile loop (clamped LDS gathers + cndmask), v_rcp for the
// normalizer division to keep phase 3 under the 23.3 TB/s HBM roofline.

typedef __attribute__((ext_vector_type(2))) float v2f;
typedef __attribute__((ext_vector_type(4))) float v4f;
typedef __attribute__((ext_vector_type(8))) float v8f;

#define T_LEN   64000
#define CH_N    512      // B*C = 8*64
#define L_CHUNK 512
#define N_CHUNK 125      // 64000 / 512
#define TILES   32       // 512 / 16
#define LDSP    20       // [ch][time] pad: 16B-aligned rows, conflict-free

__device__ __forceinline__ float powi(float a, int n) {
  float r = 1.0f;
  for (int i = 0; i < n; ++i) r *= a;
  return r;
}

// ---------------- Phase 1: chunk summaries -------------------------------
// S[k][ch] = sum_{i=0..511} mu * a^(511-i) * relu(x[ch][k*512+i])
// lane L handles i = 32j+L: inner recurrence ratio a^32, outer weight mu*a^(31-L).
__global__ __launch_bounds__(256) void p1_summaries(
    const float* __restrict__ x, const float* __restrict__ pmuf,
    const float* __restrict__ pmus, float* __restrict__ Sf,
    float* __restrict__ Ss) {
  const float muf = *pmuf, mus = *pmus;
  const float af = 1.0f - muf, as = 1.0f - mus;
  const float af32 = powi(af, 32), as32 = powi(as, 32);

  const int wid  = blockIdx.x * 8 + (threadIdx.x >> 5);  // 0..63999
  const int lane = threadIdx.x & 31;
  const int ch   = wid / N_CHUNK;
  const int k    = wid % N_CHUNK;

  const float* p = x + (size_t)ch * T_LEN + k * L_CHUNK + lane;
  float sf = 0.0f, ss = 0.0f;
#pragma unroll
  for (int j = 0; j < 16; ++j) {            // coalesced: 32 lanes x stride-32 steps
    float v = fmaxf(p[j * 32], 0.0f);
    sf = af32 * sf + v;
    ss = as32 * ss + v;
  }
  sf *= muf * powi(af, 31 - lane);
  ss *= mus * powi(as, 31 - lane);
#pragma unroll
  for (int off = 16; off; off >>= 1) {      // wave32 butterfly reduce
    sf += __shfl_xor(sf, off, 32);
    ss += __shfl_xor(ss, off, 32);
  }
  if (lane == 0) {
    Sf[k * CH_N + ch] = sf;
    Ss[k * CH_N + ch] = ss;
  }
}

// ---------------- Phase 2: compose carries across chunks ------------------
// m_{-1} = relu(x0) reproduces M[0]=x0 exactly (a + mu = 1).
// C[k][ch] = M value just before chunk k; m_k = a^512 * m_{k-1} + S_k.
__global__ __launch_bounds__(256) void p2_carries(
    const float* __restrict__ x, const float* __restrict__ pmuf,
    const float* __restrict__ pmus, const float* __restrict__ Sf,
    const float* __restrict__ Ss, float* __restrict__ Cf,
    float* __restrict__ Cs) {
  const int ch = blockIdx.x * 256 + threadIdx.x;  // grid=2 -> 512 threads
  const float muf = *pmuf, mus = *pmus;
  const float af = 1.0f - muf, as = 1.0f - mus;
  float aLf = powi(af, 32); aLf *= aLf; aLf *= aLf; aLf *= aLf; aLf *= aLf; // a^512
  float aLs = powi(as, 32); aLs *= aLs; aLs *= aLs; aLs *= aLs; aLs *= aLs;

  const float x0 = fmaxf(x[(size_t)ch * T_LEN], 0.0f);
  float mf = x0, ms = x0;
  for (int k = 0; k < N_CHUNK; ++k) {       // coalesced across ch
    Cf[k * CH_N + ch] = mf;
    Cs[k * CH_N + ch] = ms;
    mf = aLf * mf + Sf[k * CH_N + ch];
    ms = aLs * ms + Ss[k * CH_N + ch];
  }
}

// A-operand element: logical columns = [decay d | W lower-tri Toeplitz | pad]
// col 0        : a^(row+1)              (carry injection)
// col 1..16    : mu * a^(row-(col-1)) if row >= col-1 else 0
// col 17..19   : 0 (K padding to 20)
__device__ __forceinline__ float aval(int row, int col, float mu, float a) {
  if (col == 0) return powi(a, row + 1);
  if (col <= 16 && row >= col - 1) return mu * powi(a, row - (col - 1));
  return 0.0f;
}

// ---------------- Phase 3: WMMA reconstruction + output -------------------
// One wave handles 16 channels x one 512-sample chunk, marching 32 tiles of
// 16 timesteps. Per tile: M_tile(16t x 16ch) = A(16x20) @ [m_in; X](20x16)
// done as 5 chained v_wmma_f32_16x16x4_f32 per EMA.
__global__ __launch_bounds__(256) void p3_reconstruct(
    const float* __restrict__ x, const float* __restrict__ pmuf,
    const float* __restrict__ pmus, const float* __restrict__ pwa,
    const float* __restrict__ pwh, const float* __restrict__ Cf,
    const float* __restrict__ Cs, float* __restrict__ out) {
  __shared__ __align__(16) float xs[8][16][LDSP];  // per-wave [ch][time] tile

  const float muf = *pmuf, mus = *pmus;
  const float af = 1.0f - muf, as = 1.0f - mus;
  const float mixa = 1.0f / (1.0f + __expf(-(*pwa)));
  const float mixh = 1.0f / (1.0f + __expf(-(*pwh)));

  const int warp = threadIdx.x >> 5;
  const int lane = threadIdx.x & 31;
  const int c    = lane & 15;               // channel-in-group == D column
  const int half = lane >> 4;               // 0: D rows 0-7, 1: D rows 8-15

  const int wid = blockIdx.x * 8 + warp;    // 0..3999
  const int k   = wid >> 5;                 // chunk 0..124
  const int g   = wid & 31;                 // channel group 0..31
  const int ch  = g * 16 + c;

  // Loop-invariant A operands (A 16x4 layout: row=lane%16, VGPR0=K{0,2}, VGPR1=K{1,3}).
  v2f Af[5], As[5];
#pragma unroll
  for (int kb = 0; kb < 5; ++kb) {
    const int col0 = 4 * kb + 2 * half;
    Af[kb][0] = aval(c, col0, muf, af);
    Af[kb][1] = aval(c, col0 + 1, muf, af);
    As[kb][0] = aval(c, col0, mus, as);
    As[kb][1] = aval(c, col0 + 1, mus, as);
  }

  // Chunk carry-in (meaningful in lanes 0-15, one per channel).
  const int cidx = k * CH_N + g * 16 + c;
  float minf = Cf[cidx];
  float mins = Cs[cidx];

  float* myrow = &xs[warp][c][0];
  const float* xp = x + (size_t)ch * T_LEN + k * L_CHUNK + half * 8;
  float*       op = out + (size_t)ch * T_LEN + k * L_CHUNK + half * 8;

  for (int tile = 0; tile < TILES; ++tile) {
    // Vectorized per-lane load of 8 contiguous samples (matches C/D row layout).
    v4f xa = *(const v4f*)(xp);
    v4f xb = *(const v4f*)(xp + 4);
    __builtin_prefetch(xp + 16, 0, 1);      // global_prefetch_b8: next tile
#pragma unroll
    for (int j = 0; j < 4; ++j) { xa[j] = fmaxf(xa[j], 0.0f); xb[j] = fmaxf(xb[j], 0.0f); }

    // Stage tile to LDS as [ch][time]: two ds_store_b128 per lane (in-order DS,
    // intra-wave only, no barrier needed).
    *(v4f*)(myrow + half * 8)     = xa;
    *(v4f*)(myrow + half * 8 + 4) = xb;

    // B rows: 0 = m_in, 1..16 = x[t0 + r - 1], 17..19 = 0.
    // B 4x16 layout mirrors A: VGPR0 = rows K{0,2}, VGPR1 = rows K{1,3}.
    // Branch-free: unconditional clamped ds_load, then cndmask the value.
    v8f cf = {}; v8f cs = {};
#pragma unroll
    for (int kb = 0; kb < 5; ++kb) {
      const int ra = 4 * kb + 2 * half;     // global row for B VGPR0
      const int rb = ra + 1;                // global row for B VGPR1
      const int ia = (ra >= 1) ? ((ra <= 16) ? ra - 1 : 15) : 0;
      const int ib = (rb <= 16) ? rb - 1 : 15;
      const float va = myrow[ia];
      const float vb = myrow[ib];
      const float b0x = (ra >= 1 && ra <= 16) ? va : 0.0f;
      const float b1  = (rb <= 16) ? vb : 0.0f;
      v2f Bf, Bs;
      Bf[0] = (ra == 0) ? minf : b0x;  Bf[1] = b1;
      Bs[0] = (ra == 0) ? mins : b0x;  Bs[1] = b1;
      cf = __builtin_amdgcn_wmma_f32_16x16x4_f32(false, Af[kb], false, Bf,
                                                 (short)0, cf, false, false);
      cs = __builtin_amdgcn_wmma_f32_16x16x4_f32(false, As[kb], false, Bs,
                                                 (short)0, cs, false, false);
    }

    // Next-tile carry = D row 15 (VGPR7, lanes 16-31) broadcast to lanes 0-15.
    const float mf15 = __shfl(cf[7], c + 16, 32);
    const float ms15 = __shfl(cs[7], c + 16, 32);

    // Fused envelope mix + normalize + high-pass; rows match xa/xb exactly.
    // v_rcp_f32 (~1 ulp) instead of the IEEE div expansion: keeps VALU off
    // the critical path of this HBM-bound phase.
    v4f oa, ob;
#pragma unroll
    for (int j = 0; j < 8; ++j) {
      const float M  = mixa * cf[j] + (1.0f - mixa) * cs[j];
      const float xv = (j < 4) ? xa[j] : xb[j - 4];
      const float r  = __builtin_amdgcn_rcpf(1e-6f + M);
      const float o  = xv * r + mixh * (xv - M);
      if (j < 4) oa[j] = o; else ob[j - 4] = o;
    }
    *(v4f*)op       = oa;
    *(v4f*)(op + 4) = ob;

    minf = mf15;
    mins = ms15;
    xp += 16;
    op += 16;
  }
}

extern "C" void kernel_launch(void* const* d_in, const int* in_sizes, int n_in,
                              void* d_out, int out_size, void* d_ws, size_t ws_size,
                              hipStream_t stream) {
  (void)in_sizes; (void)n_in; (void)out_size; (void)ws_size;
  const float* x   = (const float*)d_in[0];
  const float* muf = (const float*)d_in[1];
  const float* mus = (const float*)d_in[2];
  const float* wa  = (const float*)d_in[3];
  const float* wh  = (const float*)d_in[4];
  float* out = (float*)d_out;

  float* ws = (float*)d_ws;                  // 4 planes * 64000 f32 = 1 MB
  float* Sf = ws;
  float* Ss = ws + N_CHUNK * CH_N;
  float* Cf = ws + 2 * N_CHUNK * CH_N;
  float* Cs = ws + 3 * N_CHUNK * CH_N;

  p1_summaries<<<(CH_N * N_CHUNK) / 8, 256, 0, stream>>>(x, muf, mus, Sf, Ss);
  p2_carries<<<2, 256, 0, stream>>>(x, muf, mus, Sf, Ss, Cf, Cs);
  p3_reconstruct<<<500, 256, 0, stream>>>(x, muf, mus, wa, wh, Cf, Cs, out);
}